// gat_66907000537300
// MI455X (gfx1250) — compile-verified
//
#include <hip/hip_runtime.h>
#include <math.h>

#define N_NODES 50000
#define E_EDGES 800000
#define E_TOT   (E_EDGES + N_NODES)   // 850000 edges incl. self-loops
#define IN_CH   128
#define H1      8
#define C1      32
#define F1      (H1*C1)               // 256
#define C2      64
#define NEG_SLOPE 0.2f

typedef __attribute__((ext_vector_type(2))) float v2f;
typedef __attribute__((ext_vector_type(8))) float v8f;

// ---------- helpers ----------

__device__ __forceinline__ unsigned f2ord(float f) {
    unsigned u = __float_as_uint(f);
    return (u & 0x80000000u) ? ~u : (u | 0x80000000u);
}
__device__ __forceinline__ float ord2f(unsigned u) {
    return __uint_as_float((u & 0x80000000u) ? (u & 0x7fffffffu) : ~u);
}
__device__ __forceinline__ void edge_sd(const int* __restrict__ es,
                                        const int* __restrict__ ed,
                                        int e, int& s, int& d) {
    if (e < E_EDGES) { s = es[e]; d = ed[e]; }
    else             { s = e - E_EDGES; d = s; }   // appended self-loops
}

// ---------- zero fill (grid-stride) ----------

__global__ void fill_zero(float* __restrict__ p, long n) {
    long i = (long)blockIdx.x * blockDim.x + threadIdx.x;
    long stride = (long)gridDim.x * blockDim.x;
    for (; i < n; i += stride) p[i] = 0.0f;
}

// ---------- fp32 WMMA GEMM: C[M,16*NT] = A[M,K] @ B[K,16*NT] ----------
// Compile-time N,K: fully strength-reduced addressing, 16 v_wmma per loop body.
// One wave per 16x16 output tile (M%16==0 guaranteed; EXEC stays all-ones).
// A-fragment (16x4 f32): lanes 0-15 -> M=lane, K=(0,1); lanes 16-31 -> M=lane-16, K=(2,3)
// C/D: VGPR r, lanes 0-15 -> (M=r, N=lane); lanes 16-31 -> (M=r+8, N=lane-16)
template <int N, int K>
__global__ __launch_bounds__(256) void wmma_gemm_f32(
    const float* __restrict__ A, const float* __restrict__ B,
    float* __restrict__ C, int M)
{
    int wave = (int)((blockIdx.x * (unsigned)blockDim.x + threadIdx.x) >> 5);
    int lane = threadIdx.x & 31;
    constexpr int NT = N >> 4;
    int mt = wave / NT;
    int nt = wave - mt * NT;
    if (mt >= (M >> 4)) return;            // wave-uniform: EXEC stays all-ones

    int m0 = mt << 4, n0 = nt << 4;
    int half = lane >> 4;                  // 0: K pair (0,1)   1: K pair (2,3)
    int lr   = lane & 15;

    const float* arow = A + (long)(m0 + lr) * K + 2 * half;
    const float* bcol = B + (long)(2 * half) * N + n0 + lr;

    v8f acc = {};
    #pragma unroll 16
    for (int k = 0; k < K; k += 4) {
        v2f a, b;
        a.x = arow[0];        a.y = arow[1];        // A[m, k+2h], A[m, k+2h+1]
        b.x = bcol[0];        b.y = bcol[N];        // B[k+2h, n], B[k+2h+1, n]
        acc = __builtin_amdgcn_wmma_f32_16x16x4_f32(
            /*neg_a=*/false, a, /*neg_b=*/false, b,
            /*c_mod=*/(short)0, acc, /*reuse_a=*/false, /*reuse_b=*/false);
        arow += 4;
        bcol += 4 * (long)N;
    }

    float* crow = C + (long)(m0 + 8 * half) * N + n0 + lr;
    #pragma unroll
    for (int r = 0; r < 8; r++) crow[(long)r * N] = acc[r];
}

// ---------- per-(node,head) attention logits ----------

__global__ void alpha_kernel(const float* __restrict__ h,
                             const float* __restrict__ att_src,
                             const float* __restrict__ att_dst,
                             float* __restrict__ a_src,
                             float* __restrict__ a_dst,
                             int n, int H, int C)
{
    int t = blockIdx.x * blockDim.x + threadIdx.x;
    if (t >= n * H) return;
    int node = t / H, head = t - node * H;
    const float* hp = h + (long)node * H * C + head * C;
    const float* as = att_src + head * C;
    const float* ad = att_dst + head * C;
    float s = 0.f, d = 0.f;
    for (int c = 0; c < C; c++) { float v = hp[c]; s += v * as[c]; d += v * ad[c]; }
    a_src[t] = s;
    a_dst[t] = d;
}

// ---------- segment softmax: pass 1 (max), pass 2 (exp + denom) ----------

__global__ void edge_max_kernel(const int* __restrict__ es, const int* __restrict__ ed,
                                const float* __restrict__ a_src,
                                const float* __restrict__ a_dst,
                                unsigned* __restrict__ emax, int H)
{
    long t = (long)blockIdx.x * blockDim.x + threadIdx.x;
    if (t >= (long)E_TOT * H) return;
    int e = (int)(t / H), h = (int)(t - (long)e * H);
    int s, d; edge_sd(es, ed, e, s, d);
    float x = a_src[s * H + h] + a_dst[d * H + h];
    x = (x > 0.f) ? x : NEG_SLOPE * x;
    atomicMax(&emax[d * H + h], f2ord(x));
}

__global__ void edge_sum_kernel(const int* __restrict__ es, const int* __restrict__ ed,
                                const float* __restrict__ a_src,
                                const float* __restrict__ a_dst,
                                const unsigned* __restrict__ emax,
                                float* __restrict__ p, float* __restrict__ denom, int H)
{
    long t = (long)blockIdx.x * blockDim.x + threadIdx.x;
    if (t >= (long)E_TOT * H) return;
    int e = (int)(t / H), h = (int)(t - (long)e * H);
    int s, d; edge_sd(es, ed, e, s, d);
    float x = a_src[s * H + h] + a_dst[d * H + h];
    x = (x > 0.f) ? x : NEG_SLOPE * x;
    float pe = expf(x - ord2f(emax[d * H + h]));
    p[(long)e * H + h] = pe;
    atomicAdd(&denom[d * H + h], pe);
}

// ---------- weighted scatter aggregation ----------
// One thread per (edge, 4 consecutive features): b128 gather from hmat,
// 4x GLOBAL_ATOMIC_ADD_F32 scatter. Edge/coef lookups amortized 4x.
// C must be a multiple of 4 (32 and 64 here).

__global__ void edge_aggr4_kernel(const int* __restrict__ es, const int* __restrict__ ed,
                                  const float4* __restrict__ hmat,
                                  const float* __restrict__ p,
                                  const float* __restrict__ denom,
                                  float* __restrict__ out, int H, int C)
{
    int F4 = (H * C) >> 2;
    long t = (long)blockIdx.x * blockDim.x + threadIdx.x;
    if (t >= (long)E_TOT * F4) return;
    int e = (int)(t / F4);
    int q = (int)(t - (long)e * F4);       // quad index within the feature row
    int h = q / (C >> 2);
    int s, d; edge_sd(es, ed, e, s, d);
    float coef = p[(long)e * H + h] / denom[d * H + h];
    float4 v = hmat[(long)s * F4 + q];     // global_load_b128 (L2-resident)
    float* o = out + ((long)d * F4 + q) * 4;
    atomicAdd(o + 0, v.x * coef);
    atomicAdd(o + 1, v.y * coef);
    atomicAdd(o + 2, v.z * coef);
    atomicAdd(o + 3, v.w * coef);
}

// ---------- bias + relu (layer 1 epilogue) ----------

__global__ void bias_relu_kernel(float* __restrict__ x, const float* __restrict__ b,
                                 long n, int F)
{
    long i = (long)blockIdx.x * blockDim.x + threadIdx.x;
    if (i >= n) return;
    float v = x[i] + b[(int)(i % F)];
    x[i] = v > 0.f ? v : 0.f;
}

// ---------- bias + log_softmax over 64 channels: one wave per node ----------

__global__ __launch_bounds__(256) void logsoftmax_kernel(const float* __restrict__ acc,
                                                         const float* __restrict__ bias,
                                                         float* __restrict__ out)
{
    int wave = (int)((blockIdx.x * (unsigned)blockDim.x + threadIdx.x) >> 5);
    int lane = threadIdx.x & 31;
    if (wave >= N_NODES) return;
    const float* row = acc + (long)wave * C2;
    float v0 = row[lane]      + bias[lane];
    float v1 = row[lane + 32] + bias[lane + 32];
    float m = fmaxf(v0, v1);
    #pragma unroll
    for (int off = 16; off > 0; off >>= 1) m = fmaxf(m, __shfl_xor(m, off, 32));
    float s = expf(v0 - m) + expf(v1 - m);
    #pragma unroll
    for (int off = 16; off > 0; off >>= 1) s += __shfl_xor(s, off, 32);
    float lse = m + logf(s);
    float* orow = out + (long)wave * C2;
    orow[lane]      = v0 - lse;
    orow[lane + 32] = v1 - lse;
}

// ---------- host launch ----------

extern "C" void kernel_launch(void* const* d_in, const int* in_sizes, int n_in,
                              void* d_out, int out_size, void* d_ws, size_t ws_size,
                              hipStream_t stream) {
    const float* x   = (const float*)d_in[0];
    const int*   ei  = (const int*)  d_in[1];   // [2, 800000]
    const float* W1  = (const float*)d_in[2];
    const float* as1 = (const float*)d_in[3];
    const float* ad1 = (const float*)d_in[4];
    const float* b1  = (const float*)d_in[5];
    const float* W2  = (const float*)d_in[6];
    const float* as2 = (const float*)d_in[7];
    const float* ad2 = (const float*)d_in[8];
    const float* b2  = (const float*)d_in[9];
    float* out = (float*)d_out;

    const int* es = ei;
    const int* ed = ei + E_EDGES;

    // ---- workspace layout (floats) ----
    float* ws = (float*)d_ws;
    float*    h1    = ws;                                     // N*F1 (reused by layer 2)
    float*    out1  = h1    + (long)N_NODES * F1;             // N*F1
    float*    asr1  = out1  + (long)N_NODES * F1;             // N*H1
    float*    adr1  = asr1  + (long)N_NODES * H1;             // N*H1
    unsigned* emax1 = (unsigned*)(adr1 + (long)N_NODES * H1); // N*H1
    float*    den1  = (float*)emax1 + (long)N_NODES * H1;     // N*H1
    float*    p1    = den1  + (long)N_NODES * H1;             // E_TOT*H1
    // layer-2 buffers alias the h1 region (h1 is dead after edge_aggr layer 1)
    float*    h2    = h1;                                     // N*C2
    float*    out2  = h2    + (long)N_NODES * C2;             // N*C2
    float*    asr2  = out2  + (long)N_NODES * C2;             // N
    float*    adr2  = asr2  + N_NODES;                        // N
    unsigned* emax2 = (unsigned*)(adr2 + N_NODES);            // N
    float*    den2  = (float*)emax2 + N_NODES;                // N
    float*    p2    = den2  + N_NODES;                        // E_TOT

    const int B = 256;

    // ===== Layer 1 =====
    // zero: out1 | asr1 | adr1 | emax1 | den1 (contiguous). ord-encoded 0 < any float.
    {
        long n = (long)N_NODES * F1 + 4L * N_NODES * H1;
        fill_zero<<<4096, B, 0, stream>>>(out1, n);
    }
    // h1 = x @ W1   [50000,128]@[128,256]
    {
        int tiles = (N_NODES / 16) * (F1 / 16);               // 50000 waves
        wmma_gemm_f32<F1, IN_CH><<<(tiles + 7) / 8, B, 0, stream>>>(x, W1, h1, N_NODES);
    }
    alpha_kernel<<<(N_NODES * H1 + B - 1) / B, B, 0, stream>>>(h1, as1, ad1, asr1, adr1,
                                                              N_NODES, H1, C1);
    {
        long n = (long)E_TOT * H1;
        unsigned g = (unsigned)((n + B - 1) / B);
        edge_max_kernel<<<g, B, 0, stream>>>(es, ed, asr1, adr1, emax1, H1);
        edge_sum_kernel<<<g, B, 0, stream>>>(es, ed, asr1, adr1, emax1, p1, den1, H1);
    }
    {
        long n = (long)E_TOT * (F1 / 4);                      // 54.4M threads
        unsigned g = (unsigned)((n + B - 1) / B);
        edge_aggr4_kernel<<<g, B, 0, stream>>>(es, ed, (const float4*)h1, p1, den1,
                                               out1, H1, C1);
    }
    bias_relu_kernel<<<(N_NODES * (long)F1 + B - 1) / B, B, 0, stream>>>(
        out1, b1, (long)N_NODES * F1, F1);

    // ===== Layer 2 =====
    // zero: out2 | asr2 | adr2 | emax2 | den2 (contiguous)
    {
        long n = (long)N_NODES * C2 + 4L * N_NODES;
        fill_zero<<<4096, B, 0, stream>>>(out2, n);
    }
    // h2 = relu(out1) @ W2   [50000,256]@[256,64]
    {
        int tiles = (N_NODES / 16) * (C2 / 16);               // 12500 waves
        wmma_gemm_f32<C2, F1><<<(tiles + 7) / 8, B, 0, stream>>>(out1, W2, h2, N_NODES);
    }
    alpha_kernel<<<(N_NODES + B - 1) / B, B, 0, stream>>>(h2, as2, ad2, asr2, adr2,
                                                         N_NODES, 1, C2);
    {
        unsigned g = (unsigned)((E_TOT + B - 1) / B);
        edge_max_kernel<<<g, B, 0, stream>>>(es, ed, asr2, adr2, emax2, 1);
        edge_sum_kernel<<<g, B, 0, stream>>>(es, ed, asr2, adr2, emax2, p2, den2, 1);
    }
    {
        long n = (long)E_TOT * (C2 / 4);                      // 13.6M threads
        unsigned g = (unsigned)((n + B - 1) / B);
        edge_aggr4_kernel<<<g, B, 0, stream>>>(es, ed, (const float4*)h2, p2, den2,
                                               out2, 1, C2);
    }
    // bias + log_softmax -> d_out
    {
        int waves = N_NODES;
        logsoftmax_kernel<<<(waves * 32 + B - 1) / B, B, 0, stream>>>(out2, b2, out);
    }
}